// PartitionPadding_77257871720574
// MI455X (gfx1250) — compile-verified
//
#include <hip/hip_runtime.h>
#include <hip/hip_bf16.h>
#include <stdint.h>

// Kernel 1: starts[t] = lower_bound(ind, t) for t in [0, B]; starts[B] == N.
// Indicator is sorted, so this is the exclusive cumsum of per-molecule counts.
__global__ void pp_compute_starts(const int* __restrict__ ind, int N,
                                  const int* __restrict__ d_max_atoms,
                                  int rows, int* __restrict__ starts) {
    int t = blockIdx.x * blockDim.x + threadIdx.x;
    int ma = *d_max_atoms;                 // uniform scalar load
    int Bm = rows / ma;                    // number of molecules
    if (t > Bm) return;
    int lo = 0, hi = N;
    while (lo < hi) {
        int mid = (lo + hi) >> 1;
        if (ind[mid] < t) lo = mid + 1; else hi = mid;
    }
    starts[t] = lo;
}

// Kernel 2: each thread owns one float4 (16B) lane-slot in R=4 output rows.
// Phase 1: issue up to R async global->LDS loads (data rows) and direct
//          zero b128 stores (pad rows).
// Phase 2: one s_wait_asynccnt, then R async LDS->global stores.
// Every output byte is written exactly once; every input row read exactly once.
#define PP_R 4

__global__ void pp_scatter(const float* __restrict__ feats,
                           const int* __restrict__ starts,
                           const int* __restrict__ d_max_atoms,
                           float* __restrict__ out,
                           int rows, int D) {
    __shared__ float4 stage[256 * PP_R];   // 16 KB staging

    int tpr = D >> 2;                      // float4s (threads) per row == 64
    int rib = threadIdx.x / tpr;           // row-slice index in block (0..3)
    int q   = threadIdx.x - rib * tpr;     // float4 index within row
    int rpb = blockDim.x / tpr;            // row slices per iteration == 4
    int rowBase = blockIdx.x * (rpb * PP_R) + rib;

    int ma = *d_max_atoms;                 // uniform scalar load

    // Incremental mol/slot: one division total, then stride-wrap updates.
    int mol  = rowBase / ma;
    int slot = rowBase - mol * ma;

    // Low 32 bits of the flat LDS pointer == LDS byte offset in this wave's
    // allocation (aperture tag bits live in [63:32]).
    unsigned ldsBase = (unsigned)(uintptr_t)(&stage[0]) + threadIdx.x * 16u;

    float4*  dsts[PP_R];
    bool     isData[PP_R];

    // ---- Phase 1: issue async loads / direct zero stores ----
#pragma unroll
    for (int r = 0; r < PP_R; ++r) {
        int row = rowBase + r * rpb;
        bool active = row < rows;
        int s0 = starts[mol];
        int s1 = starts[mol + 1];
        float4* dst = (float4*)(out + (long)row * D) + q;
        bool data = active && (slot < (s1 - s0));
        dsts[r]   = dst;
        isData[r] = data;
        if (data) {
            const float4* src = (const float4*)(feats + (long)(s0 + slot) * D) + q;
            unsigned lds = ldsBase + (unsigned)(r * 256 * 16);
            asm volatile("global_load_async_to_lds_b128 %0, %1, off"
                         :: "v"(lds), "v"(src) : "memory");   // ASYNCcnt++
        } else if (active) {
            float4 z; z.x = 0.f; z.y = 0.f; z.z = 0.f; z.w = 0.f;
            *dst = z;                       // global_store_b128, STOREcnt path
        }
        // advance (mol, slot) by rpb rows (rpb <= ma for this geometry)
        slot += rpb;
        while (slot >= ma) { slot -= ma; ++mol; }
    }

    // ---- Phase 2: drain loads once, then issue async stores ----
    asm volatile("s_wait_asynccnt 0x0" ::: "memory");
#pragma unroll
    for (int r = 0; r < PP_R; ++r) {
        if (isData[r]) {
            unsigned lds = ldsBase + (unsigned)(r * 256 * 16);
            asm volatile("global_store_async_from_lds_b128 %0, %1, off"
                         :: "v"(dsts[r]), "v"(lds) : "memory"); // ASYNCcnt++
        }
    }
    // implicit S_WAIT_IDLE at s_endpgm drains outstanding async stores
}

extern "C" void kernel_launch(void* const* d_in, const int* in_sizes, int n_in,
                              void* d_out, int out_size, void* d_ws, size_t ws_size,
                              hipStream_t stream) {
    const float* feats = (const float*)d_in[0];   // [N, D] float32
    const int*   ind   = (const int*)d_in[1];     // [N]    int32 (sorted)
    const int*   d_ma  = (const int*)d_in[3];     // scalar max_atoms on device
    float*       out   = (float*)d_out;           // [B, max_atoms, D]

    int N = in_sizes[1];
    int D = in_sizes[0] / N;                      // 256
    int rows = out_size / D;                      // B * max_atoms = 262144

    int* starts = (int*)d_ws;                     // (B+1) ints, <= ~16 KB

    // 1) segment-start table (tiny; over-provisioned grid, guarded on device)
    int g1 = (rows + 1 + 255) / 256;
    pp_compute_starts<<<g1, 256, 0, stream>>>(ind, N, d_ma, rows, starts);

    // 2) single pass over output rows, R rows per thread
    int tpr = D >> 2;                             // 64
    int rpb = 256 / tpr;                          // 4
    int rowsPerBlock = rpb * PP_R;                // 16
    int g2 = (rows + rowsPerBlock - 1) / rowsPerBlock;  // 16384 blocks
    pp_scatter<<<g2, 256, 0, stream>>>(feats, starts, d_ma, out, rows, D);
}